// ConnectNet_36928128811527
// MI455X (gfx1250) — compile-verified
//
#include <hip/hip_runtime.h>

typedef float v2f __attribute__((ext_vector_type(2)));
typedef float v8f __attribute__((ext_vector_type(8)));

#define C_IN   128
#define H_IN   224
#define W_IN   224
#define C_OUT  256
#define K_TOT  (C_IN * 9)     // 1152
#define H_OUT  222
#define W_OUT  222
#define HW_OUT (H_OUT * W_OUT)

#define M_BLK  128
#define N_BLK  64
#define KT     16
#define NPANEL (K_TOT / KT)   // 72
#define A_STR  18             // padded A stride (floats): 18*m mod 64 distinct, even for b64 frags
#define PB_STR (N_BLK * 2 + 4) // 132 floats: k-pair-interleaved B row, even -> 8B-aligned pairs

__global__ __launch_bounds__(256)
void conv3x3_wmma_f32(const float* __restrict__ x,
                      const float* __restrict__ w,
                      float* __restrict__ out) {
  __shared__ float As[M_BLK * A_STR];        // A[m][k]
  __shared__ float Bs[(KT / 2) * PB_STR];    // B pair-interleaved: [(k>>1)][2n + (k&1)]

  const int tid  = threadIdx.x;
  const int lane = tid & 31;
  const int wave = tid >> 5;
  const int wm   = wave & 3;                 // M rows wm*32 .. +31
  const int wn   = wave >> 2;                // N cols wn*32 .. +31

  const int owBlk = blockIdx.x * N_BLK;      // 0,64,128,192
  const int oh    = blockIdx.y;              // 0..221
  const int mBase = blockIdx.z * M_BLK;      // 0 or 128

  v8f acc[2][2] = {};

  // A staging: 128 rows x 16 cols; 2 threads/row, 8 floats each (two float4 loads).
  const int mA = tid >> 1;                   // 0..127
  const int kA = (tid & 1) * 8;              // 0 or 8
  // B staging: 16 rows x 64 cols; thread -> (kt, 4 consecutive j).
  const int ktB = tid >> 4;                  // 0..15
  const int j0  = (tid & 15) * 4;            // 0..60

  const int half = lane >> 4;                // K sub-pair select
  const int l16  = lane & 15;

  float4 aR0, aR1;                           // staged A regs
  float  bR[4];                              // staged B regs

  // ---- global-load helper: panel at kBase -> registers only
  auto loadG = [&](int kBase) {
    const float4* src =
        (const float4*)(w + (size_t)(mBase + mA) * K_TOT + kBase + kA);
    aR0 = src[0];
    aR1 = src[1];
    const int k  = kBase + ktB;              // im2col: k = c*9 + kh*3 + kw
    const int c  = k / 9;
    const int r  = k - c * 9;
    const int kh = r / 3;
    const int kw = r - kh * 3;
    const float* bs =
        x + (size_t)c * (H_IN * W_IN) + (size_t)(oh + kh) * W_IN + owBlk + kw + j0;
#pragma unroll
    for (int i = 0; i < 4; ++i) {
      const int ow = owBlk + j0 + i;         // ow<222 => column ow+kw <= 223 in-range
      bR[i] = (ow < W_OUT) ? bs[i] : 0.0f;
    }
  };

  loadG(0);                                  // prologue: panel 0 in flight

  for (int p = 0; p < NPANEL; ++p) {
    // ---- commit staged registers to LDS
    {
      float* ad = &As[mA * A_STR + kA];
      ad[0] = aR0.x; ad[1] = aR0.y; ad[2] = aR0.z; ad[3] = aR0.w;
      ad[4] = aR1.x; ad[5] = aR1.y; ad[6] = aR1.z; ad[7] = aR1.w;
      float* bd = &Bs[(ktB >> 1) * PB_STR + j0 * 2 + (ktB & 1)];
#pragma unroll
      for (int i = 0; i < 4; ++i) bd[2 * i] = bR[i];
    }
    __syncthreads();

    // ---- kick off next panel's global loads; they complete behind the WMMAs
    if (p + 1 < NPANEL) loadG((p + 1) * KT);

    // ---- 4 K-micro-steps of 4 -> 16 v_wmma_f32_16x16x4_f32 per panel per wave
#pragma unroll
    for (int k0 = 0; k0 < KT; k0 += 4) {
      v2f a[2], b[2];
#pragma unroll
      for (int ti = 0; ti < 2; ++ti)         // A frag: K = k0+2h, k0+2h+1 (contiguous, b64)
        a[ti] = *(const v2f*)&As[(wm * 32 + ti * 16 + l16) * A_STR + k0 + 2 * half];
#pragma unroll
      for (int tj = 0; tj < 2; ++tj)         // B frag: pair row k0/2+h, one b64 per lane
        b[tj] = *(const v2f*)&Bs[(k0 / 2 + half) * PB_STR +
                                 (wn * 32 + tj * 16 + l16) * 2];
#pragma unroll
      for (int ti = 0; ti < 2; ++ti)
#pragma unroll
        for (int tj = 0; tj < 2; ++tj)
          acc[ti][tj] = __builtin_amdgcn_wmma_f32_16x16x4_f32(
              false, a[ti], false, b[tj], (short)0, acc[ti][tj], false, false);
    }
    __syncthreads();
  }

  // ---- epilogue: C/D layout -> VGPR r holds M = r + 8*half, N = l16
#pragma unroll
  for (int ti = 0; ti < 2; ++ti) {
    const int m0 = mBase + wm * 32 + ti * 16 + half * 8;
#pragma unroll
    for (int tj = 0; tj < 2; ++tj) {
      const int ow = owBlk + wn * 32 + tj * 16 + l16;
      if (ow < W_OUT) {
        float* dst = out + (size_t)m0 * HW_OUT + (size_t)oh * W_OUT + ow;
#pragma unroll
        for (int r = 0; r < 8; ++r)
          dst[(size_t)r * HW_OUT] = acc[ti][tj][r];
      }
    }
  }
}

extern "C" void kernel_launch(void* const* d_in, const int* in_sizes, int n_in,
                              void* d_out, int out_size, void* d_ws, size_t ws_size,
                              hipStream_t stream) {
  const float* x = (const float*)d_in[0];    // (128,224,224) f32
  const float* k = (const float*)d_in[1];    // (256,128,3,3) f32
  float* out = (float*)d_out;                // (256,222,222) f32

  dim3 grid((W_OUT + N_BLK - 1) / N_BLK,     // 4 ow-chunks
            H_OUT,                           // 222 rows
            C_OUT / M_BLK);                  // 2 m-chunks
  conv3x3_wmma_f32<<<grid, 256, 0, stream>>>(x, k, out);
}